// MinPooling2d_74096775790662
// MI455X (gfx1250) — compile-verified
//
#include <hip/hip_runtime.h>

// MinPooling2d: input (1, 64, 1024, 1024) f32, k=3, stride 2, VALID
// -> output (1, 64, 511, 511) f32.  Pure HBM-bound: ~323 MB of traffic
// => ~13.9 us floor at 23.3 TB/s.  No matrix structure -> no WMMA; the
// CDNA5 win is the memory path (b128 coalesced loads, register-rolled
// vertical reuse, NT stores, global_prefetch_b8).

#define C_    64
#define H_    1024
#define W_    1024
#define HOUT_ 511
#define WOUT_ 511
#define HCHUNK_  8
#define HGROUPS_ 64   // ceil(511/8)
#define WGROUPS_ 128  // ceil(511/4); last group has 3 valid columns

__device__ __forceinline__ float min3f(float a, float b, float c) {
    return fminf(fminf(a, b), c);   // -> v_min3_num_f32
}

// Horizontal 3-wide/stride-2 min over 9 consecutive floats (cols 2w0..2w0+8),
// producing 4 output-column partial mins.  r must be 16B aligned (it is:
// byte offset = 32 * wgroup).  r[8] may spill into the next row's col 0 for
// the tail w-group; that lane of the result is never stored.
__device__ __forceinline__ float4 rowmin9(const float* __restrict__ r) {
    const float4 v0 = *(const float4*)(r);
    const float4 v1 = *(const float4*)(r + 4);
    const float  s  = r[8];
    float4 h;
    h.x = min3f(v0.x, v0.y, v0.z);   // cols 2w0+0,1,2
    h.y = min3f(v0.z, v0.w, v1.x);   // cols 2w0+2,3,4
    h.z = min3f(v1.x, v1.y, v1.z);   // cols 2w0+4,5,6
    h.w = min3f(v1.z, v1.w, s);      // cols 2w0+6,7,8
    return h;
}

__global__ __launch_bounds__(256)
void MinPooling2d_74096775790662_kernel(const float* __restrict__ img,
                                        float* __restrict__ out) {
    const int t  = blockIdx.x * blockDim.x + threadIdx.x;
    const int wg = t & (WGROUPS_ - 1);          // 0..127
    const int hg = (t >> 7) & (HGROUPS_ - 1);   // 0..63
    const int c  = t >> 13;                     // 0..63
    if (c >= C_) return;

    const int  w0    = wg * 4;
    const int  h0    = hg * HCHUNK_;
    const int  nrows = min(HCHUNK_, HOUT_ - h0);     // 8, or 7 for last hg
    const bool tail  = (wg == WGROUPS_ - 1);         // only 3 output cols

    const float* __restrict__ base =
        img + ((size_t)c * H_ + (size_t)(2 * h0)) * W_ + (size_t)(2 * w0);
    float* __restrict__ obase =
        out + ((size_t)c * HOUT_ + (size_t)h0) * WOUT_ + (size_t)w0;

    // Register carry: horizontal mins of input row 2*(h0+i); each input row
    // is loaded exactly once per thread.
    float4 prev = rowmin9(base);              // row 2*h0
    const float* r = base + W_;               // row 2*h0 + 1

    for (int i = 0; i < nrows; ++i) {
        const float4 a = rowmin9(r);          // row 2*(h0+i) + 1
        const float4 b = rowmin9(r + W_);     // row 2*(h0+i) + 2
        r += 2 * W_;

        float4 o;
        o.x = min3f(prev.x, a.x, b.x);
        o.y = min3f(prev.y, a.y, b.y);
        o.z = min3f(prev.z, a.z, b.z);
        o.w = min3f(prev.w, a.w, b.w);
        prev = b;                             // carries to next output row

        // Pull a future row toward WGP$/L2 (speculative, OOB-safe).
        __builtin_prefetch(r + 2 * W_, 0, 1); // -> global_prefetch_b8

        // Wout = 511 -> rows are 4B-aligned only; scalar NT stores
        // (adjacent lanes jointly fill cachelines; keep L2 for input reuse).
        float* op = obase + (size_t)i * WOUT_;
        __builtin_nontemporal_store(o.x, op + 0);
        __builtin_nontemporal_store(o.y, op + 1);
        __builtin_nontemporal_store(o.z, op + 2);
        if (!tail) __builtin_nontemporal_store(o.w, op + 3);
    }
}

extern "C" void kernel_launch(void* const* d_in, const int* in_sizes, int n_in,
                              void* d_out, int out_size, void* d_ws, size_t ws_size,
                              hipStream_t stream) {
    (void)in_sizes; (void)n_in; (void)d_ws; (void)ws_size; (void)out_size;
    const float* img = (const float*)d_in[0];   // (1,64,1024,1024) f32
    // d_in[1] = kernel_size (int, ==3) -- shapes are compile-time constants here.
    float* out = (float*)d_out;                 // (1,64,511,511) f32, fully written

    const int total  = C_ * HGROUPS_ * WGROUPS_;   // 524288 threads
    const int block  = 256;                        // 8 wave32 waves
    const int grid   = total / block;              // 2048, exact
    MinPooling2d_74096775790662_kernel<<<grid, block, 0, stream>>>(img, out);
}